// SamplingSeedActor_90640989815328
// MI455X (gfx1250) — compile-verified
//
#include <hip/hip_runtime.h>

// ---------------------------------------------------------------------------
// SamplingSeedActor: binary-decode z -> seed, add obs_hash, mod max_seed,
// gather 256-float rows from seed_to_action. Pure HBM-bound gather-copy
// (~32 MB total traffic -> ~1.4us at 23.3 TB/s on MI455X). Wave-per-row;
// seed decode is wave-uniform (scalar loads + SALU), row copy uses the
// gfx1250 async global<->LDS engines (ASYNCcnt) with B128 transfers.
// ---------------------------------------------------------------------------

#define GPTR __attribute__((address_space(1)))
#define LPTR __attribute__((address_space(3)))

#ifndef __has_builtin
#define __has_builtin(x) 0
#endif

#if __has_builtin(__builtin_amdgcn_global_load_async_to_lds_b128) && \
    __has_builtin(__builtin_amdgcn_global_store_async_from_lds_b128) && \
    __has_builtin(__builtin_amdgcn_s_wait_asynccnt)
#define USE_ASYNC_LDS 1
#else
#define USE_ASYNC_LDS 0
#endif

typedef int v4i __attribute__((ext_vector_type(4)));

namespace {
constexpr int kZDim = 18;                      // bits per z row
constexpr int kADim = 256;                     // floats per action row (1 KB)
constexpr int kWavesPerBlock = 8;              // 256 threads / wave32
constexpr int kDefaultMaxSeed = (1 << 18) + 20000;  // 282144 (reference MAX_SEED)
}  // namespace

// kConstMaxSeed > 0: modulo by compile-time constant (mul+shift, no SALU div
// expansion). kConstMaxSeed == 0: runtime modulo fallback.
template <int kConstMaxSeed>
__global__ __launch_bounds__(256) void seed_gather_kernel(
    const int* __restrict__ obs_hash,   // [batch]
    const int* __restrict__ z,          // [batch, 18]
    const float* __restrict__ table,    // [max_seed, 256]
    float* __restrict__ out,            // [batch, 256]
    int batch, int max_seed_rt) {
  const int lane = threadIdx.x & 31;
  const int wave = __builtin_amdgcn_readfirstlane((int)(threadIdx.x >> 5));
  const int row  = (int)blockIdx.x * kWavesPerBlock + wave;
  if (row >= batch) return;  // wave-uniform exit

  // Wave-uniform seed decode: row is uniform, so these lower to scalar loads.
  const int* zr = z + row * kZDim;
  int seed = 0;
#pragma unroll
  for (int k = 0; k < kZDim; ++k) seed = (seed << 1) | zr[k];
  const int max_seed = (kConstMaxSeed > 0) ? kConstMaxSeed : max_seed_rt;
  seed = (seed + obs_hash[row]) % max_seed;          // sum < 2^21, i32 exact
  seed = __builtin_amdgcn_readfirstlane(seed);       // force SGPR gather base

  const float* src = table + (size_t)seed * kADim;
  float* dst = out + (size_t)row * kADim;

#if USE_ASYNC_LDS
  // Route the 1 KB row through LDS with the async copy engines: two B128
  // async loads per lane (offsets 0 and 512), wait ASYNCcnt, two B128 async
  // stores per lane. Per-wave LDS slice; no cross-wave sharing -> no barrier.
  __shared__ __align__(16) float lds_buf[kWavesPerBlock * kADim];  // 8 KB
  float* wbuf = lds_buf + wave * kADim;

  const int off = lane * 16;
  GPTR v4i* gsrc4 = (GPTR v4i*)((GPTR char*)src + off);   // const dropped
  GPTR v4i* gdst4 = (GPTR v4i*)((GPTR char*)dst + off);
  LPTR v4i* lds4  = (LPTR v4i*)((LPTR char*)wbuf + off);

  __builtin_amdgcn_global_load_async_to_lds_b128(gsrc4, lds4, /*offset=*/0,   /*cpol=*/0);
  __builtin_amdgcn_global_load_async_to_lds_b128(gsrc4, lds4, /*offset=*/512, /*cpol=*/0);

  __builtin_amdgcn_s_wait_asynccnt(0);  // LDS row valid

  __builtin_amdgcn_global_store_async_from_lds_b128(gdst4, lds4, /*offset=*/0,   /*cpol=*/0);
  __builtin_amdgcn_global_store_async_from_lds_b128(gdst4, lds4, /*offset=*/512, /*cpol=*/0);
  // s_endpgm performs an implicit wait-idle, draining the async stores.
#else
  // Fallback: direct B128 register copy, 2 x float4 per lane.
  const float4* s4 = (const float4*)src;
  float4* d4 = (float4*)dst;
  float4 a = s4[lane];
  float4 b = s4[lane + 32];
  d4[lane]      = a;
  d4[lane + 32] = b;
#endif
}

extern "C" void kernel_launch(void* const* d_in, const int* in_sizes, int n_in,
                              void* d_out, int out_size, void* d_ws, size_t ws_size,
                              hipStream_t stream) {
  const int*   obs_hash = (const int*)d_in[0];     // [batch] int
  const int*   z        = (const int*)d_in[1];     // [batch*18] int
  const float* table    = (const float*)d_in[2];   // [max_seed*256] f32
  float*       out      = (float*)d_out;           // [batch*256] f32

  const int batch    = in_sizes[0];
  const int max_seed = in_sizes[2] / kADim;
  const int blocks   = (batch + kWavesPerBlock - 1) / kWavesPerBlock;

  if (max_seed == kDefaultMaxSeed) {
    seed_gather_kernel<kDefaultMaxSeed><<<dim3(blocks), dim3(256), 0, stream>>>(
        obs_hash, z, table, out, batch, max_seed);
  } else {
    seed_gather_kernel<0><<<dim3(blocks), dim3(256), 0, stream>>>(
        obs_hash, z, table, out, batch, max_seed);
  }
}